// SimpleRNNClassifier_5488968204860
// MI455X (gfx1250) — compile-verified
//
#include <hip/hip_runtime.h>
#include <cmath>

// ---------------------------------------------------------------------------
// SimpleRNNClassifier for MI455X (gfx1250, wave32, WMMA)
//   B=512, T=2048, I=16, H=40, L=3; fp32 throughout (matches reference).
//   Matrix op: V_WMMA_F32_16X16X4_F32 (16x16 f32 D = 16x4 A * 4x16 B + C).
//
// Strategy:
//   * per layer: fully-parallel WMMA projection GEMM over all B*T rows,
//     then a latency-bound sequential scan (32 blocks x 3 waves) with the
//     hidden state in LDS and W_hh^T fragments pinned in registers.
//   * weights/bias are pre-packed zero-padded (H=40 -> 48) so every fragment
//     load is an unconditional 8-byte load: no exec masking, no movrels.
// ---------------------------------------------------------------------------

typedef __attribute__((ext_vector_type(2))) float f32x2;
typedef __attribute__((ext_vector_type(8))) float f32x8;

#define RNN_B 512
#define RNN_T 2048
#define RNN_I 16
#define RNN_H 40
#define PADH  48   // H padded to 3 WMMA tiles of 16
#define LDSW  50   // LDS row stride (floats): even -> 8B-aligned b64 loads,
                   // 50*m mod 64 hits 16 distinct banks for m = 0..15

static __device__ __forceinline__ f32x8 wmma_f32_16x16x4(f32x2 a, f32x2 b, f32x8 c) {
  // 8-arg form: (neg_a, A, neg_b, B, c_mod, C, reuse_a, reuse_b)
  return __builtin_amdgcn_wmma_f32_16x16x4_f32(false, a, false, b, (short)0, c,
                                               false, false);
}

#if __has_builtin(__builtin_amdgcn_tanhf)
static __device__ __forceinline__ float fast_tanh(float x) {
  return __builtin_amdgcn_tanhf(x);           // V_TANH_F32 (gfx1250 TRANS op)
}
#else
static __device__ __forceinline__ float fast_tanh(float x) { return tanhf(x); }
#endif

// ---------------------------------------------------------------------------
// Pack one layer's weights into zero-padded 48x48 row-major [n][k] buffers and
// fuse the two bias vectors. Runs once per layer; removes all bounds checks
// from the hot kernels.
// ---------------------------------------------------------------------------
__global__ __launch_bounds__(256)
void rnn_pack_kernel(const float* __restrict__ wih, int in_dim,
                     const float* __restrict__ whh,
                     const float* __restrict__ bih,
                     const float* __restrict__ bhh,
                     float* __restrict__ wih_pad,
                     float* __restrict__ whh_pad,
                     float* __restrict__ bias_pad)
{
  const int idx = blockIdx.x * blockDim.x + threadIdx.x;
  if (idx < PADH * PADH) {
    const int n = idx / PADH;
    const int k = idx % PADH;
    wih_pad[idx] = (n < RNN_H && k < in_dim) ? wih[n * in_dim + k] : 0.0f;
    whh_pad[idx] = (n < RNN_H && k < RNN_H)  ? whh[n * RNN_H  + k] : 0.0f;
  }
  if (idx < PADH)
    bias_pad[idx] = (idx < RNN_H) ? (bih[idx] + bhh[idx]) : 0.0f;
}

// ---------------------------------------------------------------------------
// Projection GEMM over all B*T rows:
//   out[row, 0:48] = in[row, :] @ Wpad^T + bias_pad       (pad lanes stay 0)
// Each wave owns one 16-row M-tile; 8 waves per block; KCHUNKS is a compile-
// time constant (4 for layer 0, 12 for layers 1-2) so everything unrolls and
// the A fragments live in statically addressed VGPRs.
// f32 WMMA A layout: lane L -> M = L%16; VGPR0/1 hold K = kb+koff, kb+koff+1
// with koff = (L<16 ? 0 : 2). B layout mirrors with N = L%16.
// ---------------------------------------------------------------------------
template <int KCHUNKS>
__global__ __launch_bounds__(256)
void rnn_proj_kernel(const float* __restrict__ in, int in_stride,
                     const float* __restrict__ wpad,
                     const float* __restrict__ bias_pad,
                     float* __restrict__ outp)
{
  const int lane = threadIdx.x & 31;
  const int wave = threadIdx.x >> 5;
  const int tile = blockIdx.x * 8 + wave;          // 16-row tile index of B*T
  const int m    = lane & 15;
  const int koff = (lane < 16) ? 0 : 2;
  const int Mhi  = (lane < 16) ? 0 : 8;
  const size_t rowA = (size_t)tile * 16 + m;
  const float* arow = in + rowA * (size_t)in_stride;

  // A fragments for this wave's 16 rows (held across all three N-tiles).
  f32x2 afr[KCHUNKS];
  #pragma unroll
  for (int kc = 0; kc < KCHUNKS; ++kc)
    afr[kc] = *(const f32x2*)&arow[kc * 4 + koff];

  const size_t outbase = (size_t)tile * 16;
  #pragma unroll
  for (int nt = 0; nt < 3; ++nt) {
    const int n = nt * 16 + m;
    const float bias = bias_pad[n];
    f32x8 acc = {bias, bias, bias, bias, bias, bias, bias, bias};

    const float* brow = wpad + n * PADH;           // B[k][n] = Wpad[n][k]
    #pragma unroll
    for (int kc = 0; kc < KCHUNKS; ++kc) {
      const f32x2 bf = *(const f32x2*)&brow[kc * 4 + koff];
      acc = wmma_f32_16x16x4(afr[kc], bf, acc);
    }

    #pragma unroll
    for (int r = 0; r < 8; ++r) {
      const int M = r + Mhi;                       // C/D layout: M = r (+8 hi)
      outp[(outbase + M) * PADH + n] = acc[r];
    }
  }
}

// ---------------------------------------------------------------------------
// Sequential scan: h_t = tanh(xp_t + h_{t-1} @ Whh^T).
// One block per 16-batch tile; 3 waves, one per 16-wide N-tile of PADH=48.
// h_{t-1} lives in LDS; Whh^T fragments are pinned in registers. Per step:
// 12 chained f32 WMMAs (K=48) + hardware tanh on the serial critical path.
// ---------------------------------------------------------------------------
__global__ __launch_bounds__(96)
void rnn_scan_kernel(const float* __restrict__ xp, float* __restrict__ hseq,
                     const float* __restrict__ whh_pad)
{
  __shared__ float hbuf[16 * LDSW];

  const int lane = threadIdx.x & 31;
  const int nt   = threadIdx.x >> 5;               // 0..2: this wave's N-tile
  const int m    = lane & 15;
  const int koff = (lane < 16) ? 0 : 2;
  const int Mhi  = (lane < 16) ? 0 : 8;
  const int n    = nt * 16 + m;
  const size_t brow = (size_t)blockIdx.x * 16;

  // Recurrent weights, B[k][n] = Whh_pad[n][k]; unconditional b64 loads.
  f32x2 bfrag[12];
  {
    const float* wrow = whh_pad + n * PADH;
    #pragma unroll
    for (int kc = 0; kc < 12; ++kc)
      bfrag[kc] = *(const f32x2*)&wrow[kc * 4 + koff];
  }

  // h_0 = 0
  for (int i = threadIdx.x; i < 16 * LDSW; i += 96) hbuf[i] = 0.0f;
  __syncthreads();

  for (int t = 0; t < RNN_T; ++t) {
    // C := xp[b_tile, t, n-tile]  (C/D layout: VGPR r -> M = r (+8 hi), N = m)
    f32x8 acc;
    #pragma unroll
    for (int r = 0; r < 8; ++r) {
      const size_t row = brow + r + Mhi;
      acc[r] = xp[(row * RNN_T + t) * PADH + n];
    }

    // Prefetch xp a few steps ahead (global_prefetch_b8).
    const int tpf = (t + 8 < RNN_T) ? (t + 8) : t;
    __builtin_prefetch(&xp[((brow + m) * RNN_T + tpf) * PADH + n], 0, 1);

    // acc += h_{t-1} @ Whh^T, K = 48 in 12 chunks of 4.
    #pragma unroll
    for (int kc = 0; kc < 12; ++kc) {
      const f32x2 a = *(const f32x2*)&hbuf[m * LDSW + kc * 4 + koff];
      acc = wmma_f32_16x16x4(a, bfrag[kc], acc);
    }

    // tanh; pad columns (n >= 40) see 0-input and 0-weights so tanh(0)=0,
    // keeping the padding exactly zero without any masking.
    float hv[8];
    #pragma unroll
    for (int r = 0; r < 8; ++r) hv[r] = fast_tanh(acc[r]);

    __syncthreads();                               // all waves done reading h_{t-1}
    #pragma unroll
    for (int r = 0; r < 8; ++r) {
      const int M = r + Mhi;
      hbuf[M * LDSW + n] = hv[r];
      hseq[((brow + M) * RNN_T + t) * PADH + n] = hv[r];
    }
    __syncthreads();                               // h_t visible to all waves
  }
}

// ---------------------------------------------------------------------------
// Final FC: out[b] = h_last[b, :H] . fc_w + fc_b
// ---------------------------------------------------------------------------
__global__ __launch_bounds__(256)
void rnn_fc_kernel(const float* __restrict__ hseq, const float* __restrict__ fcw,
                   const float* __restrict__ fcb, float* __restrict__ out)
{
  const int b = blockIdx.x * blockDim.x + threadIdx.x;
  if (b >= RNN_B) return;
  const float* h = hseq + ((size_t)b * RNN_T + (RNN_T - 1)) * PADH;
  float s = fcb[0];
  #pragma unroll
  for (int i = 0; i < RNN_H; ++i) s += h[i] * fcw[i];
  out[b] = s;
}

// ---------------------------------------------------------------------------
// Host-side launch. Inputs (setup_inputs order):
//  0:x  1:w_ih0 2:w_hh0 3:b_ih0 4:b_hh0  5..8: layer1  9..12: layer2
//  13:fc_w 14:fc_b
// Workspace: 2 ping-pong seq buffers (201 MB each) + ~56 KB packed weights.
// ---------------------------------------------------------------------------
extern "C" void kernel_launch(void* const* d_in, const int* in_sizes, int n_in,
                              void* d_out, int out_size, void* d_ws, size_t ws_size,
                              hipStream_t stream)
{
  (void)in_sizes; (void)n_in; (void)out_size; (void)ws_size;

  const float* x      = (const float*)d_in[0];
  const float* wih[3] = {(const float*)d_in[1], (const float*)d_in[5], (const float*)d_in[9]};
  const float* whh[3] = {(const float*)d_in[2], (const float*)d_in[6], (const float*)d_in[10]};
  const float* bih[3] = {(const float*)d_in[3], (const float*)d_in[7], (const float*)d_in[11]};
  const float* bhh[3] = {(const float*)d_in[4], (const float*)d_in[8], (const float*)d_in[12]};
  const float* fcw    = (const float*)d_in[13];
  const float* fcb    = (const float*)d_in[14];
  float*       out    = (float*)d_out;

  const size_t seq_elems = (size_t)RNN_B * RNN_T * PADH;
  float* bufA = (float*)d_ws;          // xp buffer
  float* bufB = bufA + seq_elems;      // h-sequence buffer
  float* pads = bufB + seq_elems;      // packed weights/bias, per layer:
  const size_t perLayer = (size_t)PADH * PADH * 2 + PADH;

  // Pack all three layers (tiny kernels, removes bounds checks from hot path).
  for (int l = 0; l < 3; ++l) {
    float* wp = pads + l * perLayer;
    rnn_pack_kernel<<<dim3((PADH * PADH + 255) / 256), dim3(256), 0, stream>>>(
        wih[l], (l == 0) ? RNN_I : RNN_H, whh[l], bih[l], bhh[l],
        wp, wp + PADH * PADH, wp + 2 * PADH * PADH);
  }

  const int rows = RNN_B * RNN_T;      // 1,048,576
  dim3 projGrid(rows / 128), projBlk(256);   // 8 waves * 16 rows per block
  dim3 scanGrid(RNN_B / 16), scanBlk(96);    // 3 waves (one per N-tile)

  for (int l = 0; l < 3; ++l) {
    float* wp  = pads + l * perLayer;
    float* whp = wp + PADH * PADH;
    float* bp  = wp + 2 * PADH * PADH;
    if (l == 0)
      rnn_proj_kernel<RNN_I / 4><<<projGrid, projBlk, 0, stream>>>(
          x, RNN_I, wp, bp, bufA);
    else
      rnn_proj_kernel<PADH / 4><<<projGrid, projBlk, 0, stream>>>(
          bufB, PADH, wp, bp, bufA);
    rnn_scan_kernel<<<scanGrid, scanBlk, 0, stream>>>(bufA, bufB, whp);
  }
  rnn_fc_kernel<<<dim3((RNN_B + 255) / 256), dim3(256), 0, stream>>>(
      bufB, fcw, fcb, out);
}